// OutputBlock_65901978189931
// MI455X (gfx1250) — compile-verified
//
#include <hip/hip_runtime.h>

#define NFEAT 128
#define NHID  64
#define NRAD  6

// contiguous 16-edge tiles processed by each wave in the edge kernel
#define EDGE_TILES_PER_WAVE 4

typedef __attribute__((ext_vector_type(2))) float v2f;
typedef __attribute__((ext_vector_type(8))) float v8f;

__device__ __forceinline__ float swishf(float x) {
    // x * sigmoid(x)
    return x / (1.0f + __expf(-x));
}

__device__ __forceinline__ void atomAddF32(float* p, float v) {
    __hip_atomic_fetch_add(p, v, __ATOMIC_RELAXED, __HIP_MEMORY_SCOPE_AGENT);
}

// ---------------------------------------------------------------------------
// Zero the atom_feat accumulator (N*128 floats) in workspace.
// ---------------------------------------------------------------------------
__global__ void k_zero(float4* __restrict__ buf, int n4) {
    int i = blockIdx.x * blockDim.x + threadIdx.x;
    if (i < n4) buf[i] = make_float4(0.f, 0.f, 0.f, 0.f);
}

// ---------------------------------------------------------------------------
// Edge kernel: rbf_feat = rbf @ W_rbf.T via V_WMMA_F32_16X16X4_F32,
// then scatter-add (m * rbf_feat) into atom_feat with fp32 atomics.
// One wave = EDGE_TILES_PER_WAVE contiguous tiles of 16 edges.
// K=6 padded to 8 -> 2 WMMA k-steps per feature tile (branchless pad).
// A (16x4 f32): lane L holds row M=L%16; vgpr0=K(off), vgpr1=K(off+1),
//               off = 0 for lanes 0-15, 2 for lanes 16-31.
// B (4x16 f32): lane L holds col N=L%16 with the same K striping.
// D (16x16 f32): vgpr r -> row M = r + (L<16?0:8), col N = L%16.
// m loads are batched two rows (16 loads) at a time into registers so the
// compiler can clause them instead of a per-element load->wait->atomic chain.
// ---------------------------------------------------------------------------
__global__ void __launch_bounds__(256)
k_edge_wmma(const float* __restrict__ m,
            const float* __restrict__ rbf,
            const int*   __restrict__ eidx,
            const float* __restrict__ Wrbf,   // [128,6] row-major
            float*       __restrict__ atom_feat,
            int ntiles) {
    const int lane  = threadIdx.x & 31;
    const int wave  = threadIdx.x >> 5;
    const int gwave = blockIdx.x * (blockDim.x >> 5) + wave;
    const int t0    = gwave * EDGE_TILES_PER_WAVE;
    if (t0 >= ntiles) return;            // wave-uniform: EXEC stays all-ones
    int tend = t0 + EDGE_TILES_PER_WAVE;
    if (tend > ntiles) tend = ntiles;

    const int   Nl    = lane & 15;
    const int   off   = (lane < 16) ? 0 : 2;
    const int   half8 = (lane < 16) ? 0 : 8;
    const float kmask = (lane < 16) ? 1.0f : 0.0f;   // zero-pad K=6,7

    // Hoisted B fragments: W_rbf.T, tile-invariant (16 v2f = 32 VGPRs)
    v2f B0[8], B1[8];
#pragma unroll
    for (int n = 0; n < 8; ++n) {
        const float* wr = Wrbf + (n * 16 + Nl) * NRAD;
        B0[n][0] = wr[off];
        B0[n][1] = wr[off + 1];
        B1[n][0] = wr[4] * kmask;
        B1[n][1] = wr[5] * kmask;
    }

    for (int tile = t0; tile < tend; ++tile) {
        const int e0 = tile * 16;

        // A fragments: rbf rows, branchless K pad (32-bit offsets)
        const unsigned roff = (unsigned)(e0 + Nl) * NRAD;
        v2f a0, a1;
        a0[0] = rbf[roff + off];
        a0[1] = rbf[roff + off + 1];
        a1[0] = rbf[roff + 4] * kmask;
        a1[1] = rbf[roff + 5] * kmask;

        // 8 feature tiles of 16 -> rbf_feat[16 edges][128 feats] in registers
        v8f d[8];
#pragma unroll
        for (int n = 0; n < 8; ++n) {
            v8f c = {};
            c = __builtin_amdgcn_wmma_f32_16x16x4_f32(false, a0, false, B0[n],
                                                      (short)0, c, false, false);
            c = __builtin_amdgcn_wmma_f32_16x16x4_f32(false, a1, false, B1[n],
                                                      (short)0, c, false, false);
            d[n] = c;
        }

        // Multiply by m and scatter-add into atom_feat (L2-resident, 51 MB).
        // Two rows per iteration: 16 independent m loads -> one clause/wait.
#pragma unroll
        for (int rp = 0; rp < 4; ++rp) {
            const int r0 = rp * 2;
            const int r1 = rp * 2 + 1;
            const int eA = e0 + r0 + half8;
            const int eB = e0 + r1 + half8;
            const unsigned mA = (unsigned)eA * NFEAT + Nl;
            const unsigned mB = (unsigned)eB * NFEAT + Nl;

            float mvA[8], mvB[8];
#pragma unroll
            for (int n = 0; n < 8; ++n) mvA[n] = m[mA + n * 16];
#pragma unroll
            for (int n = 0; n < 8; ++n) mvB[n] = m[mB + n * 16];

            const unsigned aA = (unsigned)eidx[eA] * NFEAT + Nl;
            const unsigned aB = (unsigned)eidx[eB] * NFEAT + Nl;
#pragma unroll
            for (int n = 0; n < 8; ++n)
                atomAddF32(&atom_feat[aA + n * 16], d[n][r0] * mvA[n]);
#pragma unroll
            for (int n = 0; n < 8; ++n)
                atomAddF32(&atom_feat[aB + n * 16], d[n][r1] * mvB[n]);
        }
    }
}

// Scalar tail for E % 16 edges (not launched for E = 1.6M).
__global__ void k_edge_tail(const float* __restrict__ m,
                            const float* __restrict__ rbf,
                            const int*   __restrict__ eidx,
                            const float* __restrict__ Wrbf,
                            float*       __restrict__ atom_feat,
                            int estart, int E) {
    int t = blockIdx.x * blockDim.x + threadIdx.x;
    int e = estart + t / NFEAT;
    int f = t % NFEAT;
    if (e >= E) return;
    float s = 0.f;
#pragma unroll
    for (int k = 0; k < NRAD; ++k)
        s += rbf[(size_t)e * NRAD + k] * Wrbf[f * NRAD + k];
    atomAddF32(&atom_feat[(size_t)eidx[e] * NFEAT + f],
               s * m[(size_t)e * NFEAT + f]);
}

// ---------------------------------------------------------------------------
// Atom kernel: out = swish(swish(atom_feat) @ W1.T + b1) @ W2.T + b2
// One wave = 16 atoms. h[16,64] via 32 k-steps x 4 output tiles of
// V_WMMA_F32_16X16X4_F32, W1 staged in LDS (32 KB / block).
// A-fragment loads are batched in groups of 8 k-steps for load clustering.
// ---------------------------------------------------------------------------
__global__ void __launch_bounds__(256)
k_atom_wmma(const float* __restrict__ atom_feat,
            const float* __restrict__ W1,   // [64,128] row-major
            const float* __restrict__ b1,   // [64]
            const float* __restrict__ W2,   // [64]
            const float* __restrict__ b2,   // [1]
            float*       __restrict__ out,
            int ntiles) {
    __shared__ float ldsW1[NHID * NFEAT];   // 32 KB
    {
        const float4* src = (const float4*)W1;
        float4*       dst = (float4*)ldsW1;
#pragma unroll
        for (int j = 0; j < (NHID * NFEAT / 4) / 256; ++j)
            dst[threadIdx.x + j * 256] = src[threadIdx.x + j * 256];
    }
    __syncthreads();

    const int lane  = threadIdx.x & 31;
    const int wave  = threadIdx.x >> 5;
    const int tile  = blockIdx.x * (blockDim.x >> 5) + wave;
    if (tile >= ntiles) return;          // wave-uniform
    const int a0    = tile * 16;
    const int Nl    = lane & 15;
    const int off   = (lane < 16) ? 0 : 2;
    const int half8 = (lane < 16) ? 0 : 8;

    const unsigned xoff = (unsigned)(a0 + Nl) * NFEAT;

    v8f acc[4] = {};
#pragma unroll
    for (int g = 0; g < 4; ++g) {
        // Batch-load 8 A fragments (one clause), then swish, then WMMA.
        v2f av[8];
#pragma unroll
        for (int j = 0; j < 8; ++j) {
            const int kk = g * 8 + j;
            av[j][0] = atom_feat[xoff + 4 * kk + off];
            av[j][1] = atom_feat[xoff + 4 * kk + off + 1];
        }
#pragma unroll
        for (int j = 0; j < 8; ++j) {
            av[j][0] = swishf(av[j][0]);
            av[j][1] = swishf(av[j][1]);
        }
#pragma unroll
        for (int j = 0; j < 8; ++j) {
            const int kk = g * 8 + j;
#pragma unroll
            for (int n = 0; n < 4; ++n) {
                const v2f b =
                    *(const v2f*)&ldsW1[(n * 16 + Nl) * NFEAT + 4 * kk + off];
                acc[n] = __builtin_amdgcn_wmma_f32_16x16x4_f32(
                    false, av[j], false, b, (short)0, acc[n], false, false);
            }
        }
    }

    // Epilogue: +b1, swish, dot with W2, +b2
    const float b2v = b2[0];
    float p[8] = {0.f, 0.f, 0.f, 0.f, 0.f, 0.f, 0.f, 0.f};
#pragma unroll
    for (int n = 0; n < 4; ++n) {
        const float b1v = b1[n * 16 + Nl];
        const float w2v = W2[n * 16 + Nl];
#pragma unroll
        for (int r = 0; r < 8; ++r) {
            p[r] += swishf(acc[n][r] + b1v) * w2v;
        }
    }
    // Reduce across the 16-lane half holding each atom's columns
#pragma unroll
    for (int r = 0; r < 8; ++r) {
        float v = p[r];
        v += __shfl_xor(v, 1, 16);
        v += __shfl_xor(v, 2, 16);
        v += __shfl_xor(v, 4, 16);
        v += __shfl_xor(v, 8, 16);
        if ((lane & 15) == 0) out[a0 + r + half8] = v + b2v;
    }
}

// Scalar tail for N % 16 atoms (not launched for N = 100000).
__global__ void k_atom_tail(const float* __restrict__ atom_feat,
                            const float* __restrict__ W1,
                            const float* __restrict__ b1,
                            const float* __restrict__ W2,
                            const float* __restrict__ b2,
                            float* __restrict__ out,
                            int astart, int N) {
    int a = astart + blockIdx.x * blockDim.x + threadIdx.x;
    if (a >= N) return;
    float x[NFEAT];
#pragma unroll
    for (int f = 0; f < NFEAT; ++f)
        x[f] = swishf(atom_feat[(size_t)a * NFEAT + f]);
    float o = 0.f;
    for (int h = 0; h < NHID; ++h) {
        float s = b1[h];
#pragma unroll
        for (int f = 0; f < NFEAT; ++f)
            s += x[f] * W1[h * NFEAT + f];
        o += swishf(s) * W2[h];
    }
    out[a] = o + b2[0];
}

// ---------------------------------------------------------------------------
extern "C" void kernel_launch(void* const* d_in, const int* in_sizes, int n_in,
                              void* d_out, int out_size, void* d_ws, size_t ws_size,
                              hipStream_t stream) {
    const float* m    = (const float*)d_in[0];
    const float* rbf  = (const float*)d_in[1];
    const int*   eidx = (const int*)d_in[2];
    // d_in[3] = num_atoms scalar on device; N is derivable from out_size.
    const float* Wrbf = (const float*)d_in[4];
    const float* W1   = (const float*)d_in[5];
    const float* b1   = (const float*)d_in[6];
    const float* W2   = (const float*)d_in[7];
    const float* b2   = (const float*)d_in[8];
    float* out = (float*)d_out;

    const int E = in_sizes[0] / NFEAT;
    const int N = out_size;               // out is [N,1] float32
    float* atom_feat = (float*)d_ws;      // N*128 floats = 51.2 MB

    // 1) zero accumulator
    const int n4 = (N * NFEAT) / 4;
    k_zero<<<(n4 + 255) / 256, 256, 0, stream>>>((float4*)atom_feat, n4);

    // 2) edge projection + scatter
    const int etiles = E / 16;
    if (etiles > 0) {
        const int nwaves  = (etiles + EDGE_TILES_PER_WAVE - 1) / EDGE_TILES_PER_WAVE;
        const int nblocks = (nwaves + 7) / 8;           // 8 waves / 256-thread block
        k_edge_wmma<<<nblocks, 256, 0, stream>>>(
            m, rbf, eidx, Wrbf, atom_feat, etiles);
    }
    const int erem = E - etiles * 16;
    if (erem > 0) {
        const int tt = erem * NFEAT;
        k_edge_tail<<<(tt + 255) / 256, 256, 0, stream>>>(
            m, rbf, eidx, Wrbf, atom_feat, etiles * 16, E);
    }

    // 3) per-atom MLP
    const int atiles = N / 16;
    if (atiles > 0)
        k_atom_wmma<<<(atiles + 7) / 8, 256, 0, stream>>>(
            atom_feat, W1, b1, W2, b2, out, atiles);
    const int arem = N - atiles * 16;
    if (arem > 0)
        k_atom_tail<<<(arem + 63) / 64, 64, 0, stream>>>(
            atom_feat, W1, b1, W2, b2, out, atiles * 16, N);
}